// Conv2d_58145267253807
// MI455X (gfx1250) — compile-verified
//
#include <hip/hip_runtime.h>

typedef float v2f __attribute__((ext_vector_type(2)));
typedef float v8f __attribute__((ext_vector_type(8)));

#define NB   8
#define CI   256
#define CO   256
#define H    64
#define W    64
#define YH   129
#define YW   129
#define OH_  128
#define OW_  128

// ---------------------------------------------------------------------------
// Kernel 0: repack weights w[co][ci][3][3] -> wt[tap][ci][co] so that WMMA
// A-fragment loads are coalesced (lanes vary co contiguously).
// ---------------------------------------------------------------------------
__global__ __launch_bounds__(256) void pack_weights(
    const float* __restrict__ w, float* __restrict__ wt)
{
    const int co  = threadIdx.x;          // 0..255
    const int zi  = blockIdx.x;           // tap*256 + ci
    const int tap = zi >> 8;
    const int ci  = zi & 255;
    wt[((size_t)tap * CI + ci) * CO + co] = w[((size_t)co * CI + ci) * 9 + tap];
}

// ---------------------------------------------------------------------------
// Kernel 1: factor-2 transposed 3x3 conv as implicit GEMM on
// V_WMMA_F32_16X16X4_F32. Phase decomposition: y[2i+ph, 2j+pw] only sees taps
// with kh%2==ph, kw%2==pw. Block = 512 threads = 16 waves; the block's x
// window (2 rows x 17 cols x 256 ci, zero-padded) is staged in LDS once and
// shared by all 16 co-tiles. Inner loop: 2 coalesced global A loads +
// 2 conflict-free ds B loads + 1 WMMA, no masking (EXEC all-ones).
// ---------------------------------------------------------------------------
__global__ __launch_bounds__(512) void upconv_wmma_f32(
    const float* __restrict__ x,      // [8,256,64,64]
    const float* __restrict__ wt,     // [9,256,256] packed weights
    float* __restrict__ y)            // [8,256,129,129] (workspace)
{
    const int lane = threadIdx.x & 31;
    const int wave = threadIdx.x >> 5;    // 0..15 -> co tile
    const int t  = blockIdx.x;            // 0..8 : spatial tile (5 even, 4 odd phase)
    const int oh = blockIdx.y;            // 0..128
    const int n  = blockIdx.z;            // batch
    const int co0 = wave * 16;

    int pw, j0;
    if (t < 5) { pw = 0; j0 = t * 16; } else { pw = 1; j0 = (t - 5) * 16; }
    const int ph = oh & 1;
    const int ii = oh >> 1;
    const int nj = pw ? 64 : 65;          // valid spatial positions in this phase row

    // LDS x stage: xs[r][ci][c], c is a 17-wide column window (stride 20 pads
    // the two half-wave ci planes 40 dwords apart -> disjoint LDS bank ranges).
    __shared__ float xs[2][CI][20];
    {
        const int iwbase = j0 - 1 + pw;   // staged col c maps to iw = iwbase + c
        for (int idx = threadIdx.x; idx < 2 * 17 * CI; idx += 512) {
            const int ci  = idx / 34;
            const int rem = idx - ci * 34;
            const int r   = rem / 17;
            const int c   = rem - r * 17;
            const int ih  = ii - 1 + r + ph;   // ph=0: rows ii-1, ii ; ph=1: ii, (ii+1 unused)
            const int iw  = iwbase + c;
            float v = 0.0f;
            if (ih >= 0 && ih < H && iw >= 0 && iw < W)
                v = x[(((size_t)n * CI + ci) * H + ih) * W + iw];
            xs[r][ci][c] = v;
        }
    }
    __syncthreads();

    const int nn    = lane & 15;          // A: M index (co) | B/D: N index
    const int khalf = lane >> 4;          // K-pair select; D row offset
    const int jj    = j0 + nn;            // spatial col (ow = 2*jj + pw)

    v8f acc = {};

    int khs[2]; int nkh;
    if (ph == 0) { khs[0] = 0; khs[1] = 2; nkh = 2; } else { khs[0] = 1; nkh = 1; }
    int kws[2]; int nkw;
    if (pw == 0) { kws[0] = 0; kws[1] = 2; nkw = 2; } else { kws[0] = 1; nkw = 1; }

    for (int a = 0; a < nkh; ++a) {
        const int kh   = khs[a];
        const int rsel = ph ? 0 : (kh >> 1);
        for (int b = 0; b < nkw; ++b) {
            const int kw = kws[b];
            const int cb = pw ? 0 : (kw >> 1);    // staged col base for this tap

            const float* wtap = wt + (size_t)(kh * 3 + kw) * (CI * CO) + co0 + nn;
            const float* bsrc = &xs[rsel][0][cb + nn];

            #pragma unroll 4
            for (int ci0 = 0; ci0 < CI; ci0 += 4) {
                const int k0 = ci0 + khalf * 2;
                v2f av, bv;
                // A (16x4 f32): lane m = lane%16, VGPR v holds K = v + 2*(lane/16)
                av.x = wtap[(size_t)(k0 + 0) * CO];
                av.y = wtap[(size_t)(k0 + 1) * CO];
                // B (4x16 f32): lane n = lane%16, VGPR v holds K = v + 2*(lane/16)
                bv.x = bsrc[(k0 + 0) * 20];
                bv.y = bsrc[(k0 + 1) * 20];
                acc = __builtin_amdgcn_wmma_f32_16x16x4_f32(
                        /*neg_a=*/false, av, /*neg_b=*/false, bv,
                        /*c_mod=*/(short)0, acc, /*reuse_a=*/false, /*reuse_b=*/false);
            }
        }
    }

    // D (16x16 f32, 8 VGPRs): vgpr r -> M = r + 8*(lane/16), N = lane%16
    if (jj < nj) {
        const int ow = 2 * jj + pw;
        float* ybase = y + (((size_t)n * CO + co0) * YH + oh) * YW + ow;
        #pragma unroll
        for (int r = 0; r < 8; ++r) {
            const int m = r + 8 * khalf;
            ybase[(size_t)m * (YH * YW)] = acc[r];
        }
    }
}

// ---------------------------------------------------------------------------
// Kernel 2: depthwise 4x4 FIR (outer(1,3,3,1)^2 / 16, symmetric) + bias.
// out[oy,ox] = bias + sum_{u,v} fir[u][v] * y[oy-1+u, ox-1+v]   (zero-padded)
// 32x8 output tile per block, staged through LDS.
// ---------------------------------------------------------------------------
__global__ __launch_bounds__(256) void fir_bias(
    const float* __restrict__ y,      // [8,256,129,129]
    const float* __restrict__ bias,   // [256]
    float* __restrict__ out)          // [8,256,128,128]
{
    const int ox0 = blockIdx.x * 32;
    const int oy0 = blockIdx.y * 8;
    const int nc  = blockIdx.z;       // n*256 + c
    const int c   = nc & 255;
    const int tx  = threadIdx.x, ty = threadIdx.y;

    __shared__ float tile[11][36];    // rows oy0-1..oy0+9, cols ox0-1..ox0+33 (+pad)

    const float* yb = y + (size_t)nc * (YH * YW);
    for (int idx = ty * 32 + tx; idx < 11 * 35; idx += 256) {
        const int r  = idx / 35;
        const int cc = idx % 35;
        const int yy = oy0 - 1 + r;
        const int xx = ox0 - 1 + cc;
        float v = 0.0f;
        if (yy >= 0 && yy < YH && xx >= 0 && xx < YW) v = yb[(size_t)yy * YW + xx];
        tile[r][cc] = v;
    }
    __syncthreads();

    const float kk[4] = {1.0f, 3.0f, 3.0f, 1.0f};
    float s = 0.0f;
    #pragma unroll
    for (int u = 0; u < 4; ++u) {
        float rs = 0.0f;
        #pragma unroll
        for (int v = 0; v < 4; ++v) rs += kk[v] * tile[ty + u][tx + v];
        s += kk[u] * rs;
    }
    s = s * (1.0f / 16.0f) + bias[c];

    const int oy = oy0 + ty, ox = ox0 + tx;
    out[((size_t)nc * OH_ + oy) * OW_ + ox] = s;
}

// ---------------------------------------------------------------------------
extern "C" void kernel_launch(void* const* d_in, const int* in_sizes, int n_in,
                              void* d_out, int out_size, void* d_ws, size_t ws_size,
                              hipStream_t stream) {
    const float* x    = (const float*)d_in[0];   // [8,256,64,64]
    const float* w    = (const float*)d_in[1];   // [256,256,3,3]
    const float* bias = (const float*)d_in[2];   // [256]
    float* out = (float*)d_out;                  // [8,256,128,128]

    // Workspace layout: y (137 MB) then packed weights (2.4 MB).
    float* y  = (float*)d_ws;
    float* wt = y + (size_t)NB * CO * YH * YW;

    // Kernel 0: weight repack for coalesced A-fragment loads.
    pack_weights<<<dim3(9 * CI), 256, 0, stream>>>(w, wt);

    // Kernel 1: 9 spatial tiles x 129 rows x 8 batch; 16 waves/block (all co tiles).
    upconv_wmma_f32<<<dim3(9, 129, NB), 512, 0, stream>>>(x, wt, y);

    // Kernel 2: 128/32 x 128/8 tiles x (8*256) channel planes.
    fir_bias<<<dim3(4, 16, NB * CO), dim3(32, 8), 0, stream>>>(y, bias, out);
}